// BinAutoEncoder_29429115912337
// MI455X (gfx1250) — compile-verified
//
#include <hip/hip_runtime.h>
#include <cstdint>
#include <cstddef>

// ---------------------------------------------------------------------------
// BinAutoEncoder forward for MI455X (gfx1250), wave32.
// B=8192, C=1, DIM=2048, FC_IN=2048, SPARSITY=128.
// The two FC layers (137 of ~208 GFLOP) run on the matrix pipe via
// V_WMMA_F32_16X16X4_F32 (full fp32 precision), with float4-vectorized,
// double-buffered LDS staging. Convs are small-channel direct VALU kernels.
// ---------------------------------------------------------------------------

#define BATCH 8192
#define EDIM  2048   // DIM == FC_IN == 2048
#define SPARS 128
#define TK    32     // K chunk per LDS stage
#define LDSP  36     // padded row stride (36 floats = 144 B, 16B-aligned rows)

typedef __attribute__((ext_vector_type(2))) float v2f;
typedef __attribute__((ext_vector_type(8))) float v8f;

// ---------------------------------------------------------------------------
// Tiled fp32 WMMA GEMM:  Y[m,n] = act( sum_k X[m,k] * W[n,k] + bias[n] )
// W row-major [N][K] (PyTorch Linear layout), i.e. Y = X @ W^T + b.
// Block = 256 threads = 8 waves; block tile 64(M) x 64(N).
// Wave (wy,wx) owns rows [wy*32, wy*32+32) x cols [wx*16, wx*16+16) as two
// stacked 16x16 WMMA tiles sharing one B fragment. K staged via LDS in
// 32-wide chunks, double-buffered: chunk k+1 loads into registers while
// chunk k is consumed by 16 v_wmma_f32_16x16x4_f32 per wave.
// ACT: 0 = identity, 1 = tanh.
// ---------------------------------------------------------------------------
template <int ACT>
__global__ __launch_bounds__(256) void fc_wmma_kernel(
    const float* __restrict__ X, const float* __restrict__ W,
    const float* __restrict__ bias, float* __restrict__ Y,
    int Ndim, int Kdim) {
  __shared__ float As[2][64][LDSP];
  __shared__ float Bs[2][64][LDSP];

  const int t    = threadIdx.x;
  const int lane = t & 31;
  const int l15  = lane & 15;   // column (N) / row-within-tile selector
  const int hi   = lane >> 4;   // which half of the wave (K split)
  const int wv   = t >> 5;      // wave id 0..7
  const int wy   = wv >> 2;     // 0..1  (M half, 32 rows)
  const int wx   = wv & 3;      // 0..3  (N quarter, 16 cols)

  const int m0 = blockIdx.y * 64;
  const int n0 = blockIdx.x * 64;

  // Staging: each 64x32 tile = 512 float4; thread t handles float4 #t and
  // #(t+256): row qm0 (+32), cols qk0..qk0+3.  Coalesced 128B row segments.
  const int qm0 = t >> 3;        // 0..31
  const int qk0 = (t & 7) * 4;   // 0,4,...,28

  float4 ra0, ra1, rb0, rb1;

  auto loadRegs = [&](int k0) {
    const float* xa = X + (size_t)(m0 + qm0) * Kdim + (k0 + qk0);
    ra0 = *(const float4*)xa;
    ra1 = *(const float4*)(xa + (size_t)32 * Kdim);
    const float* wb = W + (size_t)(n0 + qm0) * Kdim + (k0 + qk0);
    rb0 = *(const float4*)wb;
    rb1 = *(const float4*)(wb + (size_t)32 * Kdim);
  };
  auto storeRegs = [&](int buf) {
    *(float4*)&As[buf][qm0][qk0]      = ra0;
    *(float4*)&As[buf][qm0 + 32][qk0] = ra1;
    *(float4*)&Bs[buf][qm0][qk0]      = rb0;
    *(float4*)&Bs[buf][qm0 + 32][qk0] = rb1;
  };

  v8f acc0 = {}, acc1 = {};

  auto compute = [&](int buf) {
#pragma unroll
    for (int kk = 0; kk < TK; kk += 4) {
      // A frag (16x4 f32): lane l holds M=l%16, K = kk + 2*(l/16) + j
      // B frag (4x16 f32): lane l holds N=l%16, K = kk + 2*(l/16) + j
      const int ka = kk + 2 * hi;
      v2f b, a0, a1;
      b.x  = Bs[buf][wx * 16 + l15][ka];
      b.y  = Bs[buf][wx * 16 + l15][ka + 1];
      a0.x = As[buf][wy * 32 + l15][ka];
      a0.y = As[buf][wy * 32 + l15][ka + 1];
      a1.x = As[buf][wy * 32 + 16 + l15][ka];
      a1.y = As[buf][wy * 32 + 16 + l15][ka + 1];
      acc0 = __builtin_amdgcn_wmma_f32_16x16x4_f32(
          false, a0, false, b, (short)0, acc0, false, false);
      acc1 = __builtin_amdgcn_wmma_f32_16x16x4_f32(
          false, a1, false, b, (short)0, acc1, false, false);
    }
  };

  // Software pipeline: prologue fills buf0, then overlap load(k+1)/compute(k).
  loadRegs(0);
  storeRegs(0);
  __syncthreads();
  int cur = 0;
  for (int k0 = TK; k0 < Kdim; k0 += TK) {
    loadRegs(k0);        // global -> regs (latency overlaps compute below)
    compute(cur);        // 16 WMMAs from LDS buf[cur]
    storeRegs(cur ^ 1);  // regs -> other LDS buffer
    __syncthreads();
    cur ^= 1;
  }
  compute(cur);

  // C/D layout: VGPR i -> M = i + 8*(lane/16), N = lane%16.
  const int n   = n0 + wx * 16 + l15;
  const float bv = bias[n];
#pragma unroll
  for (int i = 0; i < 8; ++i) {
    const int row0 = m0 + wy * 32 + 8 * hi + i;
    const int row1 = row0 + 16;
    float v0 = acc0[i] + bv;
    float v1 = acc1[i] + bv;
    if (ACT == 1) { v0 = tanhf(v0); v1 = tanhf(v1); }
    Y[(size_t)row0 * Ndim + n] = v0;
    Y[(size_t)row1 * Ndim + n] = v1;
  }
}

// ---------------------------------------------------------------------------
// Direct conv, stride 2, pad 1, 3x3, ReLU. Weights OIHW.
// ---------------------------------------------------------------------------
template <int CIN, int COUT, int HIN, int WIN, int HOUT, int WOUT>
__global__ __launch_bounds__(256) void conv_s2_kernel(
    const float* __restrict__ in, const float* __restrict__ w,
    const float* __restrict__ bias, float* __restrict__ out) {
  const size_t total = (size_t)BATCH * COUT * HOUT * WOUT;
  const size_t idx = (size_t)blockIdx.x * 256 + threadIdx.x;
  if (idx >= total) return;
  const int x = (int)(idx % WOUT);
  const int y = (int)((idx / WOUT) % HOUT);
  const int o = (int)((idx / ((size_t)WOUT * HOUT)) % COUT);
  const int b = (int)(idx / ((size_t)WOUT * HOUT * COUT));

  const float* ip = in + (size_t)b * CIN * HIN * WIN;
  const float* wp = w + (size_t)o * CIN * 9;
  float s = bias[o];
#pragma unroll
  for (int ky = 0; ky < 3; ++ky) {
    const int iy = y * 2 - 1 + ky;
    if ((unsigned)iy >= (unsigned)HIN) continue;
#pragma unroll
    for (int kx = 0; kx < 3; ++kx) {
      const int ix = x * 2 - 1 + kx;
      if ((unsigned)ix >= (unsigned)WIN) continue;
      const float* ic = ip + iy * WIN + ix;
      const float* wc = wp + ky * 3 + kx;
#pragma unroll 4
      for (int c = 0; c < CIN; ++c)
        s += ic[(size_t)c * HIN * WIN] * wc[c * 9];
    }
  }
  out[idx] = fmaxf(s, 0.0f);
}

// ---------------------------------------------------------------------------
// Transposed conv (PyTorch ConvTranspose2d), stride 2, pad 1, 3x3.
// Weight layout [CIN][COUT][3][3]. Gather form:
//   out[y,x] += in[iy,ix]*w[ky,kx] where y = iy*2 - 1 + ky.
// ---------------------------------------------------------------------------
template <int CIN, int COUT, int HIN, int WIN, int HOUT, int WOUT, int RELU>
__global__ __launch_bounds__(256) void convt_s2_kernel(
    const float* __restrict__ in, const float* __restrict__ w,
    const float* __restrict__ bias, float* __restrict__ out) {
  const size_t total = (size_t)BATCH * COUT * HOUT * WOUT;
  const size_t idx = (size_t)blockIdx.x * 256 + threadIdx.x;
  if (idx >= total) return;
  const int x = (int)(idx % WOUT);
  const int y = (int)((idx / WOUT) % HOUT);
  const int o = (int)((idx / ((size_t)WOUT * HOUT)) % COUT);
  const int b = (int)(idx / ((size_t)WOUT * HOUT * COUT));

  const float* ip = in + (size_t)b * CIN * HIN * WIN;
  float s = bias[o];
#pragma unroll
  for (int ky = 0; ky < 3; ++ky) {
    const int tt = y + 1 - ky;
    if (tt & 1) continue;           // (-1)&1==1 -> also skips tt<0
    const int iy = tt >> 1;
    if ((unsigned)iy >= (unsigned)HIN) continue;
#pragma unroll
    for (int kx = 0; kx < 3; ++kx) {
      const int uu = x + 1 - kx;
      if (uu & 1) continue;
      const int ix = uu >> 1;
      if ((unsigned)ix >= (unsigned)WIN) continue;
      const float* ic = ip + iy * WIN + ix;
      const float* wc = w + (size_t)o * 9 + ky * 3 + kx;
#pragma unroll 4
      for (int c = 0; c < CIN; ++c)
        s += ic[(size_t)c * HIN * WIN] * wc[(size_t)c * COUT * 9];
    }
  }
  out[idx] = RELU ? fmaxf(s, 0.0f) : s;
}

// ---------------------------------------------------------------------------
// Per-row top-SPARS binarize.  One 256-thread block per row of 2048.
// Keys = order-preserving uint transform of floats; binary-search the
// largest threshold T with count(key >= T) >= SPARS; kept entries become 1.0:
//   out = BS * mask + (1-BS) * emb
// Count reduction: wave32 shfl_xor tree + 8-entry LDS combine.
// ---------------------------------------------------------------------------
__global__ __launch_bounds__(256) void binarize_kernel(
    const float* __restrict__ emb, const float* __restrict__ bs_ptr,
    float* __restrict__ out) {
  __shared__ int wsum[8];
  const int row  = blockIdx.x;
  const int t    = threadIdx.x;
  const int lane = t & 31;
  const int wv   = t >> 5;

  float    vals[8];
  unsigned keys[8];
  const float* rp = emb + (size_t)row * EDIM;
#pragma unroll
  for (int i = 0; i < 8; ++i) {
    const float f = rp[t + 256 * i];
    vals[i] = f;
    const unsigned u = __float_as_uint(f);
    keys[i] = (u & 0x80000000u) ? ~u : (u | 0x80000000u);
  }

  unsigned lo = 0u, hi = 0xFFFFFFFFu;
  while (lo < hi) {
    const unsigned mid = lo + ((hi - lo) >> 1) + 1u;
    int cnt = 0;
#pragma unroll
    for (int i = 0; i < 8; ++i) cnt += (keys[i] >= mid) ? 1 : 0;
    // wave32 reduction
#pragma unroll
    for (int off = 16; off > 0; off >>= 1) cnt += __shfl_xor(cnt, off, 32);
    if (lane == 0) wsum[wv] = cnt;
    __syncthreads();
    int total = 0;
#pragma unroll
    for (int i = 0; i < 8; ++i) total += wsum[i];
    __syncthreads();
    if (total >= SPARS) lo = mid; else hi = mid - 1u;
  }
  const unsigned T = lo;

  const float BS   = *bs_ptr;
  const float omBS = 1.0f - BS;
  float* op = out + (size_t)row * EDIM;
#pragma unroll
  for (int i = 0; i < 8; ++i) {
    const float m = (keys[i] >= T) ? 1.0f : 0.0f;
    op[t + 256 * i] = BS * m + omBS * vals[i];
  }
}

// ---------------------------------------------------------------------------
// Deterministic two-stage MSE reduction.
// ---------------------------------------------------------------------------
__global__ __launch_bounds__(256) void loss_partial_kernel(
    const float* __restrict__ imgs, const float* __restrict__ recon,
    float* __restrict__ partial, int n) {
  __shared__ float red[256];
  const int t = threadIdx.x;
  float s = 0.0f;
  for (size_t i = (size_t)blockIdx.x * 256 + t; i < (size_t)n;
       i += (size_t)gridDim.x * 256) {
    const float d = imgs[i] - recon[i];
    s += d * d;
  }
  red[t] = s;
  __syncthreads();
  for (int st = 128; st > 0; st >>= 1) {
    if (t < st) red[t] += red[t + st];
    __syncthreads();
  }
  if (t == 0) partial[blockIdx.x] = red[0];
}

__global__ __launch_bounds__(256) void loss_final_kernel(
    const float* __restrict__ partial, float* __restrict__ out, int npart,
    float invN) {
  __shared__ float red[256];
  const int t = threadIdx.x;
  float s = 0.0f;
  for (int i = t; i < npart; i += 256) s += partial[i];
  red[t] = s;
  __syncthreads();
  for (int st = 128; st > 0; st >>= 1) {
    if (t < st) red[t] += red[t + st];
    __syncthreads();
  }
  if (t == 0) out[0] = red[0] * invN;
}

// ---------------------------------------------------------------------------
extern "C" void kernel_launch(void* const* d_in, const int* in_sizes, int n_in,
                              void* d_out, int out_size, void* d_ws,
                              size_t ws_size, hipStream_t stream) {
  (void)in_sizes; (void)n_in; (void)out_size; (void)ws_size;

  const float* imgs = (const float*)d_in[0];
  const float* bs   = (const float*)d_in[1];
  const float* c1w  = (const float*)d_in[2];
  const float* c1b  = (const float*)d_in[3];
  const float* c2w  = (const float*)d_in[4];
  const float* c2b  = (const float*)d_in[5];
  const float* c3w  = (const float*)d_in[6];
  const float* c3b  = (const float*)d_in[7];
  const float* few  = (const float*)d_in[8];
  const float* feb  = (const float*)d_in[9];
  const float* fdw  = (const float*)d_in[10];
  const float* fdb  = (const float*)d_in[11];
  const float* t1w  = (const float*)d_in[12];
  const float* t1b  = (const float*)d_in[13];
  const float* t2w  = (const float*)d_in[14];
  const float* t2b  = (const float*)d_in[15];
  const float* t3w  = (const float*)d_in[16];
  const float* t3b  = (const float*)d_in[17];

  // Workspace layout (floats), buffers reused along the pipeline (~443 MB):
  //   bufA: conv1 out [B,32,14,14]      -> later convT2 out [B,32,14,14]
  //   bufB: conv2 out [B,64,7,7]        -> later convT1 out [B,64,7,7]
  //   bufC: conv3 out / x_flat [B,2048] -> later decoder FC out [B,2048]
  //   bufD: emb [B,2048]                -> later loss partials
  float* ws   = (float*)d_ws;
  float* bufA = ws;                    // 51,380,224 floats
  float* bufB = bufA + 51380224;       // 25,690,112 floats
  float* bufC = bufB + 25690112;       // 16,777,216 floats
  float* bufD = bufC + 16777216;       // 16,777,216 floats

  float* out     = (float*)d_out;
  float* emb_bin = out;                // 16,777,216 floats
  float* loss    = out + 16777216;     // 1 float
  float* recon   = out + 16777217;     // 6,422,528 floats

  const dim3 blk(256);
  const dim3 gemm_grid(EDIM / 64, BATCH / 64);

  // ---- encoder ----
  conv_s2_kernel<1, 32, 28, 28, 14, 14>
      <<<dim3(51380224 / 256), blk, 0, stream>>>(imgs, c1w, c1b, bufA);
  conv_s2_kernel<32, 64, 14, 14, 7, 7>
      <<<dim3(25690112 / 256), blk, 0, stream>>>(bufA, c2w, c2b, bufB);
  conv_s2_kernel<64, 128, 7, 7, 4, 4>
      <<<dim3(16777216 / 256), blk, 0, stream>>>(bufB, c3w, c3b, bufC);
  fc_wmma_kernel<1><<<gemm_grid, blk, 0, stream>>>(bufC, few, feb, bufD,
                                                   EDIM, EDIM);

  // ---- binarize (writes emb_bin output region) ----
  binarize_kernel<<<dim3(BATCH), blk, 0, stream>>>(bufD, bs, emb_bin);

  // ---- decoder ----
  fc_wmma_kernel<0><<<gemm_grid, blk, 0, stream>>>(emb_bin, fdw, fdb, bufC,
                                                   EDIM, EDIM);
  convt_s2_kernel<128, 64, 4, 4, 7, 7, 1>
      <<<dim3(25690112 / 256), blk, 0, stream>>>(bufC, t1w, t1b, bufB);
  convt_s2_kernel<64, 32, 7, 7, 14, 14, 1>
      <<<dim3(51380224 / 256), blk, 0, stream>>>(bufB, t2w, t2b, bufA);
  convt_s2_kernel<32, 1, 14, 14, 28, 28, 0>
      <<<dim3(6422528 / 256), blk, 0, stream>>>(bufA, t3w, t3b, recon);

  // ---- loss ----
  loss_partial_kernel<<<dim3(1024), blk, 0, stream>>>(imgs, recon, bufD,
                                                      6422528);
  loss_final_kernel<<<dim3(1), blk, 0, stream>>>(bufD, loss, 1024,
                                                 1.0f / 6422528.0f);
}